// GConvNet_82197084111195
// MI455X (gfx1250) — compile-verified
//
#include <hip/hip_runtime.h>

#define NNODES 100000
#define NEDGES 1600000
#define NGRAPHS 512
#define HID 128
#define NCLS 10
#define ALPHA 0.01f
#define FMAXV 3.402823466e38f

typedef float v2f __attribute__((ext_vector_type(2)));
typedef float v8f __attribute__((ext_vector_type(8)));

// ---------------------------------------------------------------------------
// GEMM: C[M,128] = A[M,K] @ W[K,128] + bias ; ep==1 -> BN-eval scale + ReLU
// One wave computes a 16x16 tile via V_WMMA_F32_16X16X4_F32 (fp32 precision).
// Block = 256 threads = 8 waves = one 16-row stripe across all 8 col tiles.
// ---------------------------------------------------------------------------
__global__ __launch_bounds__(256) void gemm_wmma_f32(
    const float* __restrict__ A, const float* __restrict__ W,
    const float* __restrict__ bias, float* __restrict__ C,
    int K, int ep, const float* __restrict__ gamma,
    const float* __restrict__ beta)
{
    const int N = HID;
    const int wave = threadIdx.x >> 5;
    const int lane = threadIdx.x & 31;
    const int m    = lane & 15;
    const int half = lane >> 4;            // 0: K=k,k+1  1: K=k+2,k+3
    const int row0 = blockIdx.x * 16;
    const int col  = wave * 16 + m;        // this lane's output column

    v8f acc;
    const float bv = bias[col];
#pragma unroll
    for (int v = 0; v < 8; ++v) acc[v] = bv;

    const float* arow = A + (size_t)(row0 + m) * K + 2 * half;
    const float* wcol = W + (size_t)(2 * half) * N + col;

    for (int k = 0; k < K; k += 4) {
        v2f a = *(const v2f*)(arow + k);                 // A[m][k+2h], A[m][k+2h+1]
        v2f b;
        b.x = wcol[(size_t)k * N];                       // W[k+2h][col]
        b.y = wcol[(size_t)k * N + N];                   // W[k+2h+1][col]
        acc = __builtin_amdgcn_wmma_f32_16x16x4_f32(
            false, a, false, b, (short)0, acc, false, false);
    }

    if (ep == 1) {
        const float rs = 0.99999500003749971f;           // rsqrt(1 + 1e-5)
        const float g  = gamma[col] * rs;
        const float be = beta[col];
#pragma unroll
        for (int v = 0; v < 8; ++v) acc[v] = fmaxf(acc[v] * g + be, 0.0f);
    }

#pragma unroll
    for (int v = 0; v < 8; ++v) {
        const int row = row0 + v + 8 * half;             // C layout: half-wave = rows+8
        C[(size_t)row * N + col] = acc[v];
    }
}

// ---------------------------------------------------------------------------
// CSR build helpers
// ---------------------------------------------------------------------------
__global__ void init_meta(int* deg, int* cursor, int* gs, int* ge)
{
    int i = blockIdx.x * blockDim.x + threadIdx.x;
    if (i < NNODES) { deg[i] = 0; cursor[i] = 0; }
    if (i < NGRAPHS) { gs[i] = NNODES; ge[i] = 0; }
}

__global__ void count_deg(const int* __restrict__ dst, int* __restrict__ deg)
{
    int i = blockIdx.x * blockDim.x + threadIdx.x;
    if (i < NEDGES) atomicAdd(&deg[dst[i]], 1);
}

// single-workgroup hierarchical exclusive scan (100k ints -> negligible)
__global__ __launch_bounds__(1024) void scan_offsets(
    const int* __restrict__ deg, int* __restrict__ offs, int n)
{
    __shared__ int sm[1024];
    __shared__ int carry;
    if (threadIdx.x == 0) carry = 0;
    __syncthreads();
    for (int base = 0; base < n; base += 1024) {
        int i = base + (int)threadIdx.x;
        int v = (i < n) ? deg[i] : 0;
        sm[threadIdx.x] = v;
        __syncthreads();
        for (int off = 1; off < 1024; off <<= 1) {       // Hillis-Steele inclusive
            int t = (threadIdx.x >= (unsigned)off) ? sm[threadIdx.x - off] : 0;
            __syncthreads();
            sm[threadIdx.x] += t;
            __syncthreads();
        }
        int c = carry;
        if (i < n) offs[i] = c + sm[threadIdx.x] - v;    // exclusive
        __syncthreads();
        if (threadIdx.x == 0) carry = c + sm[1023];
        __syncthreads();
    }
    if (threadIdx.x == 0) offs[n] = carry;
}

__global__ void csr_fill(const int* __restrict__ src, const int* __restrict__ dst,
                         const int* __restrict__ offs, int* __restrict__ cursor,
                         int* __restrict__ csr)
{
    int i = blockIdx.x * blockDim.x + threadIdx.x;
    if (i < NEDGES) {
        int d = dst[i];
        int p = atomicAdd(&cursor[d], 1);
        csr[offs[d] + p] = src[i];
    }
}

__global__ void graph_bounds(const int* __restrict__ batch, int* gs, int* ge)
{
    int i = blockIdx.x * blockDim.x + threadIdx.x;
    if (i < NNODES) {
        int b = batch[i];
        atomicMin(&gs[b], i);
        atomicMax(&ge[b], i + 1);
    }
}

// ---------------------------------------------------------------------------
// Mean aggregation over incoming edges + LeakyReLU.
// One wave per node; lane owns 4 features; csr indices loaded coalesced then
// broadcast via shuffle; gathered rows are 512B coalesced (L2-resident).
// ---------------------------------------------------------------------------
__global__ __launch_bounds__(256) void agg_mean_leaky(
    const float* __restrict__ H, const int* __restrict__ offs,
    const int* __restrict__ csr, float* __restrict__ out)
{
    int node = (int)((blockIdx.x * blockDim.x + threadIdx.x) >> 5);
    if (node >= NNODES) return;
    const int lane = threadIdx.x & 31;
    const int beg = offs[node];
    const int end = offs[node + 1];
    const float* colp = H + lane * 4;
    float ax = 0.f, ay = 0.f, az = 0.f, aw = 0.f;

    for (int e0 = beg; e0 < end; e0 += 32) {
        int myidx = (e0 + lane < end) ? csr[e0 + lane] : 0;  // coalesced index load
        int cnt = end - e0; if (cnt > 32) cnt = 32;
        for (int j = 0; j < cnt; ++j) {
            int s = __shfl(myidx, j, 32);
            const float4 v = *(const float4*)(colp + (size_t)s * HID);
            ax += v.x; ay += v.y; az += v.z; aw += v.w;
        }
    }
    const float inv = 1.0f / fmaxf((float)(end - beg), 1.0f);
    ax *= inv; ay *= inv; az *= inv; aw *= inv;
    float4 r;
    r.x = ax > 0.f ? ax : ALPHA * ax;
    r.y = ay > 0.f ? ay : ALPHA * ay;
    r.z = az > 0.f ? az : ALPHA * az;
    r.w = aw > 0.f ? aw : ALPHA * aw;
    *(float4*)(out + (size_t)node * HID + lane * 4) = r;
}

__global__ void residual_add(float4* __restrict__ h, const float4* __restrict__ h2, int n)
{
    int i = blockIdx.x * blockDim.x + threadIdx.x;
    if (i < n) {
        float4 a = h[i], b = h2[i];
        a.x += b.x; a.y += b.y; a.z += b.z; a.w += b.w;
        h[i] = a;
    }
}

// ---------------------------------------------------------------------------
// Per-graph segment-max over the (virtual) concat [ht | h1 | h2 | h3].
// One block (128 thr) per graph; thread owns 4 consecutive concat features.
// ---------------------------------------------------------------------------
__global__ __launch_bounds__(128) void pool_max(
    const float* __restrict__ ht, const float* __restrict__ h1,
    const float* __restrict__ h2, const float* __restrict__ h3,
    const int* __restrict__ gs, const int* __restrict__ ge,
    float* __restrict__ pooled)
{
    const int g = blockIdx.x;
    const int f = (int)threadIdx.x * 4;                  // 0..511
    const float* srcp = (f < 128) ? ht : (f < 256) ? h1 : (f < 384) ? h2 : h3;
    const int off = f & 127;
    float mx = -FMAXV, my = -FMAXV, mz = -FMAXV, mw = -FMAXV;
    const int b = gs[g], e = ge[g];
    for (int i = b; i < e; ++i) {
        const float4 v = *(const float4*)(srcp + (size_t)i * HID + off);
        mx = fmaxf(mx, v.x); my = fmaxf(my, v.y);
        mz = fmaxf(mz, v.z); mw = fmaxf(mw, v.w);
    }
    float4 r; r.x = mx; r.y = my; r.z = mz; r.w = mw;
    *(float4*)(pooled + (size_t)g * 512 + f) = r;
}

// ---------------------------------------------------------------------------
// Final projection: out[512,10] = z[512,128] @ Wm2[128,10] + bm2
// One wave per graph, shuffle reduction.
// ---------------------------------------------------------------------------
__global__ __launch_bounds__(32) void mlp2(
    const float* __restrict__ z, const float* __restrict__ W,
    const float* __restrict__ b, float* __restrict__ out)
{
    const int g = blockIdx.x;
    const int lane = threadIdx.x;
    const float z0 = z[(size_t)g * HID + lane];
    const float z1 = z[(size_t)g * HID + lane + 32];
    const float z2 = z[(size_t)g * HID + lane + 64];
    const float z3 = z[(size_t)g * HID + lane + 96];
    for (int c = 0; c < NCLS; ++c) {
        float s = z0 * W[lane * NCLS + c]
                + z1 * W[(lane + 32) * NCLS + c]
                + z2 * W[(lane + 64) * NCLS + c]
                + z3 * W[(lane + 96) * NCLS + c];
        for (int off = 16; off; off >>= 1) s += __shfl_xor(s, off, 32);
        if (lane == 0) out[g * NCLS + c] = s + b[c];
    }
}

// ---------------------------------------------------------------------------
extern "C" void kernel_launch(void* const* d_in, const int* in_sizes, int n_in,
                              void* d_out, int out_size, void* d_ws, size_t ws_size,
                              hipStream_t stream)
{
    (void)in_sizes; (void)n_in; (void)out_size; (void)ws_size;
    const float* x     = (const float*)d_in[0];
    const int*   ei    = (const int*)d_in[1];          // [2, NEDGES]
    const int*   batch = (const int*)d_in[2];
    const float* W0 = (const float*)d_in[3];  const float* b0 = (const float*)d_in[4];
    const float* W1 = (const float*)d_in[5];  const float* b1 = (const float*)d_in[6];
    const float* W2 = (const float*)d_in[7];  const float* b2 = (const float*)d_in[8];
    const float* W3 = (const float*)d_in[9];  const float* b3 = (const float*)d_in[10];
    const float* Wm1 = (const float*)d_in[11]; const float* bm1 = (const float*)d_in[12];
    const float* gamma = (const float*)d_in[13]; const float* beta = (const float*)d_in[14];
    const float* Wm2 = (const float*)d_in[15]; const float* bm2 = (const float*)d_in[16];
    const int* esrc = ei;
    const int* edst = ei + NEDGES;

    char* ws = (char*)d_ws;
    size_t off = 0;
    auto alloc = [&](size_t bytes) -> void* {
        void* p = ws + off;
        off = (off + bytes + 255) & ~(size_t)255;
        return p;
    };
    const size_t featB = (size_t)NNODES * HID * sizeof(float);   // 51.2 MB
    float* tmpH   = (float*)alloc(featB);
    float* h      = (float*)alloc(featB);   // becomes h3 after residual
    float* h1     = (float*)alloc(featB);
    float* h2     = (float*)alloc(featB);
    float* ht     = (float*)alloc(featB);
    float* zbuf   = (float*)alloc((size_t)NGRAPHS * HID * sizeof(float));
    float* pooled = (float*)alloc((size_t)NGRAPHS * 512 * sizeof(float));
    int*   csr    = (int*)alloc((size_t)NEDGES * sizeof(int));
    int*   offs   = (int*)alloc((size_t)(NNODES + 1) * sizeof(int));
    int*   deg    = (int*)alloc((size_t)NNODES * sizeof(int));
    int*   cursor = (int*)alloc((size_t)NNODES * sizeof(int));
    int*   gs     = (int*)alloc((size_t)NGRAPHS * sizeof(int));
    int*   ge     = (int*)alloc((size_t)NGRAPHS * sizeof(int));

    const int TB = 256;
    // --- CSR build + graph bounds (once, reused by all 4 layers) ---
    init_meta<<<(NNODES + TB - 1) / TB, TB, 0, stream>>>(deg, cursor, gs, ge);
    count_deg<<<(NEDGES + TB - 1) / TB, TB, 0, stream>>>(edst, deg);
    scan_offsets<<<1, 1024, 0, stream>>>(deg, offs, NNODES);
    csr_fill<<<(NEDGES + TB - 1) / TB, TB, 0, stream>>>(esrc, edst, offs, cursor, csr);
    graph_bounds<<<(NNODES + TB - 1) / TB, TB, 0, stream>>>(batch, gs, ge);

    const int gemmGrid = NNODES / 16;                  // 6250 (exact)
    const int aggGrid  = (NNODES * 32 + TB - 1) / TB;  // one wave per node

    // layer 0
    gemm_wmma_f32<<<gemmGrid, TB, 0, stream>>>(x, W0, b0, tmpH, HID, 0, nullptr, nullptr);
    agg_mean_leaky<<<aggGrid, TB, 0, stream>>>(tmpH, offs, csr, h);
    // layer 1 (head)
    gemm_wmma_f32<<<gemmGrid, TB, 0, stream>>>(h, W1, b1, tmpH, HID, 0, nullptr, nullptr);
    agg_mean_leaky<<<aggGrid, TB, 0, stream>>>(tmpH, offs, csr, h1);
    // layer 2 (body)
    gemm_wmma_f32<<<gemmGrid, TB, 0, stream>>>(h1, W2, b2, tmpH, HID, 0, nullptr, nullptr);
    agg_mean_leaky<<<aggGrid, TB, 0, stream>>>(tmpH, offs, csr, h2);
    // residual: h <- h + h2  (h is now h3)
    const int n4 = NNODES * HID / 4;
    residual_add<<<(n4 + TB - 1) / TB, TB, 0, stream>>>((float4*)h, (const float4*)h2, n4);
    // layer 3 (tail)
    gemm_wmma_f32<<<gemmGrid, TB, 0, stream>>>(h, W3, b3, tmpH, HID, 0, nullptr, nullptr);
    agg_mean_leaky<<<aggGrid, TB, 0, stream>>>(tmpH, offs, csr, ht);

    // pooling over virtual concat [ht | h1 | h2 | h3]
    pool_max<<<NGRAPHS, 128, 0, stream>>>(ht, h1, h2, h, gs, ge, pooled);

    // MLP head: z = relu(BN(pooled @ Wm1 + bm1)) via WMMA (M=512, K=512)
    gemm_wmma_f32<<<NGRAPHS / 16, TB, 0, stream>>>(pooled, Wm1, bm1, zbuf, 512, 1, gamma, beta);
    mlp2<<<NGRAPHS, 32, 0, stream>>>(zbuf, Wm2, bm2, (float*)d_out);
}